// GaussianKernel_37443524886926
// MI455X (gfx1250) — compile-verified
//
#include <hip/hip_runtime.h>
#include <hip/hip_bf16.h>

typedef __attribute__((ext_vector_type(2))) float v2f;
typedef __attribute__((ext_vector_type(4))) float v4f;
typedef __attribute__((ext_vector_type(8))) float v8f;

#define BATCH      32
#define NPTS       1024
#define NTILE      16
#define NWAVES     4
#define BLOCK_THR  (NWAVES * 32)
#define LDS_STRIDE 1028   /* 1028 % 8 == 4 -> half-wave bank groups disjoint; */
                          /* 1028*4 B = 4112 = 16*257 -> rows stay 16B aligned */
#define EPS        1e-8f

__launch_bounds__(BLOCK_THR, 1)
__global__ void gaussian_softmax_kernel(const float* __restrict__ C,
                                        const float* __restrict__ masks,
                                        const float* __restrict__ inv_sigma,
                                        float* __restrict__ out) {
    __shared__ float tile[NTILE * LDS_STRIDE];  // unnormalized X, 16 rows x 1024
    __shared__ float rowsum[NWAVES * NTILE];
    __shared__ float invrow[NTILE];

    const int b   = blockIdx.x / (NPTS / NTILE);
    const int nt  = blockIdx.x % (NPTS / NTILE);
    const int n0  = nt * NTILE;

    const int tid  = threadIdx.x;
    const int wid  = tid >> 5;      // wave 0..3
    const int lane = tid & 31;
    const int lo   = lane & 15;     // column / row-within-half index
    const int hi   = lane >> 4;     // half-wave select

    const float is  = inv_sigma[0];
    const float is2 = is * is;

    const float* Cb    = C     + (size_t)b * NPTS * 3;
    const float* maskb = masks + (size_t)b * NPTS * NPTS;

    // ---- A fragment: 16x4 (rows n0..n0+15, K=0..3; K=3 zero-padded) ----
    // lanes 0-15: VGPR0=K0, VGPR1=K1 ; lanes 16-31: VGPR0=K2, VGPR1=K3(=0)
    v2f afrag;
    {
        const float* arow = Cb + (size_t)(n0 + lo) * 3;
        float a0 = arow[0], a1 = arow[1], a2 = arow[2];
        afrag.x = hi ? a2   : a0;
        afrag.y = hi ? 0.0f : a1;
    }

    // ---- sqn[r] = ||C[n0 + r + 8*hi]||^2  (matches C/D VGPR row layout) ----
    float sqn[8];
#pragma unroll
    for (int r = 0; r < 8; ++r) {
        const float* row = Cb + (size_t)(n0 + r + 8 * hi) * 3;
        float x = row[0], y = row[1], z = row[2];
        sqn[r] = x * x + y * y + z * z;
    }

    float acc[8];
#pragma unroll
    for (int r = 0; r < 8; ++r) acc[r] = 0.0f;

    // ---- main loop: this wave's 16 m-tiles (256 contiguous columns) ----
    for (int i = 0; i < 16; ++i) {
        const int m0 = (wid * 16 + i) * NTILE;
        const int m  = m0 + lo;

        // B fragment 4x16 (mirrors A layout) + sq_m from the same load
        const float* brow = Cb + (size_t)m * 3;
        float bx = brow[0], by = brow[1], bz = brow[2];
        float sqm = bx * bx + by * by + bz * bz;
        v2f bfrag;
        bfrag.x = hi ? bz   : bx;
        bfrag.y = hi ? 0.0f : by;

        v8f cz = {};
        v8f g = __builtin_amdgcn_wmma_f32_16x16x4_f32(
            /*neg_a=*/false, afrag, /*neg_b=*/false, bfrag,
            /*c_mod=*/(short)0, cz, /*reuse_a=*/false, /*reuse_b=*/false);

#pragma unroll
        for (int r = 0; r < 8; ++r) {
            const int n = n0 + r + 8 * hi;
            float msk = __builtin_nontemporal_load(&maskb[(size_t)n * NPTS + m]);
            float d  = sqn[r] + sqm - 2.0f * g[r];       // ||Cn - Cm||^2
            float aK = __expf(-is2 * d);                 // Gaussian adjacency
            float x  = __expf(-aK) * msk;
            tile[(r + 8 * hi) * LDS_STRIDE + m] = x;
            acc[r] += x;
        }
    }

    // ---- reduce row partials across the 16 lanes of each half-wave ----
#pragma unroll
    for (int r = 0; r < 8; ++r) {
        float v = acc[r];
        v += __shfl_xor(v, 1, 32);
        v += __shfl_xor(v, 2, 32);
        v += __shfl_xor(v, 4, 32);
        v += __shfl_xor(v, 8, 32);
        acc[r] = v;
    }
    if (lo == 0) {
#pragma unroll
        for (int r = 0; r < 8; ++r)
            rowsum[wid * NTILE + (r + 8 * hi)] = acc[r];
    }
    __syncthreads();

    if (tid < NTILE) {
        float s = rowsum[tid] + rowsum[NTILE + tid] +
                  rowsum[2 * NTILE + tid] + rowsum[3 * NTILE + tid];
        invrow[tid] = 1.0f / (s + EPS);
    }
    __syncthreads();

    // ---- normalize + write back: b128 LDS reads, non-temporal b128 stores ----
    float* outb = out + ((size_t)b * NPTS + n0) * NPTS;
#pragma unroll 1
    for (int row = 0; row < NTILE; ++row) {
        const float inv = invrow[row];
#pragma unroll
        for (int m4 = tid * 4; m4 < NPTS; m4 += BLOCK_THR * 4) {   // 2 iterations
            v4f v = *(const v4f*)&tile[row * LDS_STRIDE + m4];
            v.x *= inv; v.y *= inv; v.z *= inv; v.w *= inv;
            __builtin_nontemporal_store(v, (v4f*)&outb[(size_t)row * NPTS + m4]);
        }
    }
}

extern "C" void kernel_launch(void* const* d_in, const int* in_sizes, int n_in,
                              void* d_out, int out_size, void* d_ws, size_t ws_size,
                              hipStream_t stream) {
    const float* C         = (const float*)d_in[0];  // [32,1024,3]
    const float* masks     = (const float*)d_in[1];  // [32,1024,1024]
    const float* inv_sigma = (const float*)d_in[2];  // [1]
    float*       out       = (float*)d_out;          // [32,1024,1024]

    const int nblocks = BATCH * (NPTS / NTILE);      // 32 * 64 = 2048
    gaussian_softmax_kernel<<<nblocks, BLOCK_THR, 0, stream>>>(C, masks, inv_sigma, out);
}